// PerParticleDeterminantAntiequivarianceLayer_20031727468621
// MI455X (gfx1250) — compile-verified
//
#include <hip/hip_runtime.h>
#include <hip/hip_bf16.h>
#include <math.h>

// Dims from the reference
#define BSZ   2048
#define NSPIN 2
#define NPS   8
#define DFEAT 256
#define NION  4
#define DSP   3

typedef __attribute__((ext_vector_type(2))) float v2f;
typedef __attribute__((ext_vector_type(8))) float v8f;

// ---------------------------------------------------------------------------
// Kernel 1: P[s, row, 0:8]  = X[s,row,:] @ W[s, 0:256, :]   + bias[s,:]  (U part)
//           P[s, row, 8:16] = X[s,row,:] @ W[s, 256:512, :]              (V part)
// row = b*NPS + p  (M = 16384 per spin), K = 256, N = 16.
// One wave per 16-row M tile, V_WMMA_F32_16X16X4_F32 over K in steps of 4.
// Grid: 256 blocks x 256 threads (8 waves). Blocks [0,128) = spin 0.
// ---------------------------------------------------------------------------
__global__ __launch_bounds__(256) void ppd_gemm_wmma(
    const float* __restrict__ eq,     // (B, NSPIN*NPS, DFEAT)
    const float* __restrict__ W,      // (NSPIN, 2*DFEAT, NPS)
    const float* __restrict__ bias,   // (NSPIN, NPS)
    float* __restrict__ P)            // (NSPIN, 16384, 16)
{
    // LDS: combined B matrix (256 x 16) pair-packed:
    // Bs[((f>>1)*16 + n)*2 + (f&1)] = Bmat[f][n]
    __shared__ __align__(16) float Bs[DFEAT * 16];

    const int tid  = threadIdx.x;
    const int spin = blockIdx.x >> 7;          // 128 blocks per spin
    const int blk  = blockIdx.x & 127;

    const float* Wsp = W + spin * (2 * DFEAT * NPS);
    for (int idx = tid; idx < DFEAT * 16; idx += 256) {
        const int f = idx >> 4;
        const int n = idx & 15;
        const float v = (n < 8) ? Wsp[f * NPS + n]
                                : Wsp[(DFEAT + f) * NPS + (n - 8)];
        Bs[((f >> 1) * 16 + n) * 2 + (f & 1)] = v;
    }
    __syncthreads();

    const int wave  = tid >> 5;                // 0..7
    const int lane  = tid & 31;
    const int half  = lane >> 4;               // K-half select for A/B frags
    const int l15   = lane & 15;
    const int mtile = blk * 8 + wave;          // 0..1023

    // A row for this lane (A frag: lanes 0-15 K=0,1; lanes 16-31 K=2,3; M=lane&15)
    const int row    = mtile * 16 + l15;           // row within this spin
    const int bidx   = row >> 3;                   // batch index
    const int p      = row & 7;                    // electron within spin
    const float* Arow = eq + ((size_t)(bidx * (NSPIN * NPS) + spin * NPS + p)) * DFEAT;

    // Accumulator init: fold bias into the U columns (n < 8)
    const float binit = (l15 < 8) ? bias[spin * NPS + l15] : 0.0f;
    v8f c;
#pragma unroll
    for (int r = 0; r < 8; ++r) c[r] = binit;

#pragma unroll 8
    for (int kk = 0; kk < DFEAT; kk += 4) {
        const v2f a = *(const v2f*)(Arow + kk + 2 * half);
        const v2f bfrag = *(const v2f*)(&Bs[((kk >> 1) + half) * 32 + l15 * 2]);
        c = __builtin_amdgcn_wmma_f32_16x16x4_f32(
                /*neg_a=*/false, a, /*neg_b=*/false, bfrag,
                /*c_mod=*/(short)0, c, /*reuse_a=*/false, /*reuse_b=*/false);
    }

    // D layout: VGPR r -> tile row (r + 8*half), col n = l15
    float* Pout = P + ((size_t)spin * 16384 + (size_t)mtile * 16) * 16;
#pragma unroll
    for (int r = 0; r < 8; ++r) {
        Pout[(size_t)(r + 8 * half) * 16 + l15] = c[r];
    }
}

// ---------------------------------------------------------------------------
// Kernel 2: env[b,s,j,k] = sum_a env_ion[s,a,k] * exp(-|| r[b,s,j,a,:] @ M[s,k,a] ||)
// One thread per (b,s,j,k) = 262144 threads.
// ---------------------------------------------------------------------------
__global__ __launch_bounds__(256) void ppd_env(
    const float* __restrict__ r_ei,     // (B, NSPIN*NPS, NION, DSP)
    const float* __restrict__ env_dim,  // (NSPIN, NPS, NION, DSP, DSP)
    const float* __restrict__ env_ion,  // (NSPIN, NION, NPS)
    float* __restrict__ env_ws)         // (B, NSPIN, NPS(j), NPS(k))
{
    const int idx = blockIdx.x * blockDim.x + threadIdx.x;
    if (idx >= BSZ * NSPIN * NPS * NPS) return;

    const int k = idx & 7;
    const int j = (idx >> 3) & 7;
    const int s = (idx >> 6) & 1;
    const int b = idx >> 7;

    const float* rp    = r_ei   + (size_t)((b * (NSPIN * NPS) + s * NPS + j) * NION) * DSP;
    const float* Mbase = env_dim + (size_t)((s * NPS + k) * NION) * (DSP * DSP);

    float acc = 0.0f;
#pragma unroll
    for (int a = 0; a < NION; ++a) {
        const float r0 = rp[a * 3 + 0];
        const float r1 = rp[a * 3 + 1];
        const float r2 = rp[a * 3 + 2];
        const float* M = Mbase + a * 9;      // M[d][e] = M[d*3+e]
        const float e0 = r0 * M[0] + r1 * M[3] + r2 * M[6];
        const float e1 = r0 * M[1] + r1 * M[4] + r2 * M[7];
        const float e2 = r0 * M[2] + r1 * M[5] + r2 * M[8];
        const float dist = sqrtf(e0 * e0 + e1 * e1 + e2 * e2);
        acc += env_ion[(s * NION + a) * NPS + k] * expf(-dist);
    }
    env_ws[idx] = acc;
}

// ---------------------------------------------------------------------------
// Kernel 3: per-(b,s,i) 8x8 determinant of
//   Mm[j][k] = (U[b,s,i,k] + V[b,s,j,k]) * env[b,s,j,k]
// (bias already folded into U by kernel 1). Fully unrolled LU, no pivoting:
// constant indices keep the 64-element matrix in VGPRs.
// ---------------------------------------------------------------------------
__global__ __launch_bounds__(256) void ppd_det(
    const float* __restrict__ P,        // (NSPIN, 16384, 16)
    const float* __restrict__ env_ws,   // (B, NSPIN, 8, 8)
    float* __restrict__ out)            // (NSPIN, B, NPS) flat: out[s*16384 + b*8 + i]
{
    const int idx = blockIdx.x * blockDim.x + threadIdx.x;
    if (idx >= BSZ * NSPIN * NPS) return;

    const int i = idx & 7;
    const int s = (idx >> 3) & 1;
    const int b = idx >> 4;

    const float* Pu = P + ((size_t)s * 16384 + (size_t)b * 8 + i) * 16;
    float U[8];
#pragma unroll
    for (int k = 0; k < 8; ++k) U[k] = Pu[k];

    float m[8][8];
#pragma unroll
    for (int j = 0; j < 8; ++j) {
        const float* Pv = P + ((size_t)s * 16384 + (size_t)b * 8 + j) * 16 + 8;
        const float* ev = env_ws + (size_t)(((b * NSPIN + s) * NPS + j)) * NPS;
#pragma unroll
        for (int k = 0; k < 8; ++k) {
            m[j][k] = (U[k] + Pv[k]) * ev[k];
        }
    }

    float det = 1.0f;
#pragma unroll
    for (int c = 0; c < 8; ++c) {
        const float piv = m[c][c];
        det *= piv;
        const float inv = 1.0f / piv;
#pragma unroll
        for (int r = c + 1; r < 8; ++r) {
            const float f = m[r][c] * inv;
#pragma unroll
            for (int cc = c + 1; cc < 8; ++cc) {
                m[r][cc] -= f * m[c][cc];
            }
        }
    }

    out[s * (BSZ * NPS) + b * NPS + i] = det;
}

// ---------------------------------------------------------------------------
extern "C" void kernel_launch(void* const* d_in, const int* in_sizes, int n_in,
                              void* d_out, int out_size, void* d_ws, size_t ws_size,
                              hipStream_t stream) {
    const float* eq      = (const float*)d_in[0];  // (B, 16, 256)
    const float* r_ei    = (const float*)d_in[1];  // (B, 16, 4, 3)
    const float* W       = (const float*)d_in[2];  // (2, 512, 8)
    const float* bias    = (const float*)d_in[3];  // (2, 8)
    const float* env_dim = (const float*)d_in[4];  // (2, 8, 4, 3, 3)
    const float* env_ion = (const float*)d_in[5];  // (2, 4, 8)
    float* out = (float*)d_out;                    // 2 * 2048 * 8 floats

    // Workspace layout
    float* P      = (float*)d_ws;                          // 2*16384*16 = 524288 f32 (2 MB)
    float* env_ws = P + (size_t)NSPIN * 16384 * 16;        // 2048*2*8*8 = 262144 f32 (1 MB)

    // 1) WMMA GEMM: U|V projections, bias folded in
    ppd_gemm_wmma<<<256, 256, 0, stream>>>(eq, W, bias, P);

    // 2) Envelopes
    {
        const int n = BSZ * NSPIN * NPS * NPS;             // 262144
        ppd_env<<<(n + 255) / 256, 256, 0, stream>>>(r_ei, env_dim, env_ion, env_ws);
    }

    // 3) Determinants
    {
        const int n = BSZ * NSPIN * NPS;                   // 32768
        ppd_det<<<(n + 255) / 256, 256, 0, stream>>>(P, env_ws, out);
    }
}